// ParticleI2cCell_30279519437595
// MI455X (gfx1250) — compile-verified
//
#include <hip/hip_runtime.h>
#include <math.h>

typedef __attribute__((ext_vector_type(2))) float v2f;
typedef __attribute__((ext_vector_type(8))) float v8f;

#define NP 8192
#define NDX 4
#define NDU 2

// Combine two (max, sumA, sumB) logsumexp partials (log2 domain).
static __device__ __forceinline__ void lse2_combine3(float& mx, float& sA, float& sB,
                                                     float om, float osA, float osB) {
    float nm = fmaxf(mx, om);
    float r0 = __builtin_amdgcn_exp2f(mx - nm);
    float r1 = __builtin_amdgcn_exp2f(om - nm);
    sA = sA * r0 + osA * r1;
    sB = sB * r0 + osB * r1;
    mx = nm;
}

// Per-particle prologue.
//  mbuf[j][0..3] : m * (1/ln2)           (WMMA A-side, pre-scaled to log2 domain)
//  sbuf[j][0..3] : s                     (WMMA B-side, raw)
//  hm2[j]        : 0.5*|m|^2 * (1/ln2)   (row centering term; constants cancel)
//  cbuf[j]       : { exp(lw_j - 0.5|s_j|^2), exp(w_j - 0.5|s_j|^2) }
__global__ __launch_bounds__(256) void pre_kernel(
    const float* __restrict__ particles, const float* __restrict__ samples,
    const float* __restrict__ weights,   const float* __restrict__ log_weights,
    const float* __restrict__ Amat, const float* __restrict__ Bmat,
    const float* __restrict__ log_sigma,
    float* __restrict__ mbuf, float* __restrict__ sbuf,
    float* __restrict__ hm2,  float2* __restrict__ cbuf)
{
    int i = blockIdx.x * blockDim.x + threadIdx.x;
    if (i >= NP) return;
    const float INV_LN2 = 1.4426950408889634f;

    float inv_sig[NDX];
    #pragma unroll
    for (int r = 0; r < NDX; ++r) inv_sig[r] = __expf(-log_sigma[r]);

    float x[NDX], u[NDU];
    #pragma unroll
    for (int r = 0; r < NDX; ++r) x[r] = particles[i * (NDX + NDU) + r];
    #pragma unroll
    for (int r = 0; r < NDU; ++r) u[r] = particles[i * (NDX + NDU) + NDX + r];

    float nm2 = 0.0f, ns2 = 0.0f;
    #pragma unroll
    for (int r = 0; r < NDX; ++r) {
        float mean = 0.0f;
        #pragma unroll
        for (int c = 0; c < NDX; ++c) mean = __builtin_fmaf(Amat[r * NDX + c], x[c], mean);
        #pragma unroll
        for (int c = 0; c < NDU; ++c) mean = __builtin_fmaf(Bmat[r * NDU + c], u[c], mean);
        float m = mean * inv_sig[r];
        float s = samples[i * NDX + r] * inv_sig[r];
        nm2 = __builtin_fmaf(m, m, nm2);
        ns2 = __builtin_fmaf(s, s, ns2);
        mbuf[i * 4 + r] = m * INV_LN2;
        sbuf[i * 4 + r] = s;
    }
    hm2[i] = 0.5f * nm2 * INV_LN2;
    float hs = 0.5f * ns2;
    cbuf[i] = make_float2(__expf(log_weights[i] - hs), __expf(weights[i] - hs));
}

// Fused pairwise kernel: one V_WMMA_F32_16X16X4_F32 per 16x16 tile, then a
// single-transcendental dual logsumexp accumulation per element.
// Block = 128 threads (4 waves); block owns one 16-row tile; each wave strides
// over 128 of the 512 column tiles.
__global__ __launch_bounds__(128) void smooth_kernel(
    const float* __restrict__ mbuf, const float* __restrict__ sbuf,
    const float* __restrict__ hm2,  const float2* __restrict__ cbuf,
    float* __restrict__ out)
{
    const float LN2 = 0.6931471805599453f;
    __shared__ float4 partial[4][16];

    int tid  = threadIdx.x;
    int wave = tid >> 5;
    int lane = tid & 31;
    int lc   = lane & 15;   // A: row-in-tile; B/C: column-in-tile
    int half = lane >> 4;
    int k0   = half * 2;    // K pair held by this half-wave

    int rowbase = blockIdx.x * 16;

    // A fragment: A[M = lc][K = k0 + v]  (16x4 f32 layout)
    v2f afrag;
    afrag.x = mbuf[(rowbase + lc) * 4 + k0];
    afrag.y = mbuf[(rowbase + lc) * 4 + k0 + 1];

    // C/D layout: lane holds rows M = r + 8*half, column N = lc
    float hmr[8];
    #pragma unroll
    for (int r = 0; r < 8; ++r) hmr[r] = hm2[rowbase + r + 8 * half];

    float maxL[8], sumA[8], sumB[8];
    #pragma unroll
    for (int r = 0; r < 8; ++r) { maxL[r] = -INFINITY; sumA[r] = 0.0f; sumB[r] = 0.0f; }

    #pragma unroll 2
    for (int ct = wave; ct < NP / 16; ct += 4) {
        int j = ct * 16 + lc;
        // B fragment: B[K = k0 + v][N = lc] = s[j][k0 + v]
        v2f bfrag;
        bfrag.x = sbuf[j * 4 + k0];
        bfrag.y = sbuf[j * 4 + k0 + 1];
        float2 cw = cbuf[j];           // (elw', ew')

        v8f c = {};
        c = __builtin_amdgcn_wmma_f32_16x16x4_f32(
                false, afrag, false, bfrag, (short)0, c, false, false);

        #pragma unroll
        for (int r = 0; r < 8; ++r) {
            float g = c[r] - hmr[r];                         // log2-domain, centered
            float d = g - maxL[r];
            float t = __builtin_amdgcn_exp2f(-__builtin_fabsf(d));  // bare v_exp_f32
            bool up = d > 0.0f;
            float a1 = __builtin_fmaf(sumA[r], t, cw.x);     // rescale old, add new
            float a2 = __builtin_fmaf(t, cw.x, sumA[r]);     // add scaled new
            float b1 = __builtin_fmaf(sumB[r], t, cw.y);
            float b2 = __builtin_fmaf(t, cw.y, sumB[r]);
            sumA[r] = up ? a1 : a2;
            sumB[r] = up ? b1 : b2;
            maxL[r] = up ? g : maxL[r];
        }
    }

    // Reduce across the 16 lanes of each half (tile columns).
    #pragma unroll
    for (int mask = 1; mask < 16; mask <<= 1) {
        #pragma unroll
        for (int r = 0; r < 8; ++r) {
            float om  = __shfl_xor(maxL[r], mask, 32);
            float osA = __shfl_xor(sumA[r], mask, 32);
            float osB = __shfl_xor(sumB[r], mask, 32);
            lse2_combine3(maxL[r], sumA[r], sumB[r], om, osA, osB);
        }
    }

    // One lane per half publishes its 8 rows' wave-partials.
    if (lc == 0) {
        #pragma unroll
        for (int r = 0; r < 8; ++r)
            partial[wave][r + 8 * half] = make_float4(maxL[r], sumA[r], sumB[r], 0.0f);
    }
    __syncthreads();

    // Combine the 4 wave-partials per row; maxL cancels in the difference.
    if (tid < 16) {
        float4 p = partial[0][tid];
        float mL = p.x, sA = p.y, sB = p.z;
        #pragma unroll
        for (int w = 1; w < 4; ++w) {
            float4 q = partial[w][tid];
            lse2_combine3(mL, sA, sB, q.x, q.y, q.z);
        }
        out[rowbase + tid] = LN2 * (log2f(sB) - log2f(sA));
    }
}

extern "C" void kernel_launch(void* const* d_in, const int* in_sizes, int n_in,
                              void* d_out, int out_size, void* d_ws, size_t ws_size,
                              hipStream_t stream) {
    const float* particles   = (const float*)d_in[0];
    const float* samples     = (const float*)d_in[1];
    const float* weights     = (const float*)d_in[2];
    const float* log_weights = (const float*)d_in[3];
    const float* Amat        = (const float*)d_in[4];
    const float* Bmat        = (const float*)d_in[5];
    const float* log_sigma   = (const float*)d_in[6];

    float*  mbuf = (float*)d_ws;        // NP*4
    float*  sbuf = mbuf + NP * 4;       // NP*4
    float*  hm2  = sbuf + NP * 4;       // NP
    float2* cbuf = (float2*)(hm2 + NP); // NP*2  => 360,448 bytes total

    pre_kernel<<<NP / 256, 256, 0, stream>>>(particles, samples, weights, log_weights,
                                             Amat, Bmat, log_sigma,
                                             mbuf, sbuf, hm2, cbuf);
    smooth_kernel<<<NP / 16, 128, 0, stream>>>(mbuf, sbuf, hm2, cbuf, (float*)d_out);
}